// sample_79963701117627
// MI455X (gfx1250) — compile-verified
//
#include <hip/hip_runtime.h>

// Top-k masked softmax over rows of (8,4,2048,2048) fp32.
// One 256-thread workgroup (8 wave32) per row. Bandwidth-bound:
// 1.07 GB total traffic / 23.3 TB/s ~= 46 us floor.
//
// CDNA5-specific paths used (verified in disasm):
//  - global_load_async_to_lds_b128 + s_wait_asynccnt (async engine stages the row)
//  - wave32 sizing (8 per-wave privatized LDS histograms)
//  - ds_max_u32 / ds_add_u32 LDS atomics, b128 LDS/global accesses, NT stores

#define NCOL 2048
#define TPB  256
#define EPT  8   // elements per thread

typedef float v4f __attribute__((ext_vector_type(4)));

// Monotonic float -> uint key (larger float => larger key)
__device__ __forceinline__ unsigned f2key(float f) {
  unsigned u = __float_as_uint(f);
  return (u & 0x80000000u) ? ~u : (u | 0x80000000u);
}
__device__ __forceinline__ float key2f(unsigned k) {
  unsigned u = (k & 0x80000000u) ? (k & 0x7FFFFFFFu) : ~k;
  return __uint_as_float(u);
}

__global__ __launch_bounds__(TPB) void topk_softmax_kernel(
    const float* __restrict__ att, float* __restrict__ out) {
  __shared__ float    sdata[NCOL];       // staged row (8 KB)
  __shared__ unsigned shist[8][256];     // per-wave histogram copies (8 KB)
  __shared__ unsigned sred[256];         // reduced histogram
  __shared__ float    sredf[256];        // softmax-sum tree reduction
  __shared__ unsigned s_prefix;          // radix-select prefix (becomes k-th key)
  __shared__ unsigned s_remk;            // remaining rank within current bucket
  __shared__ unsigned s_maxkey;          // row max (as key)

  const int tid = threadIdx.x;
  const int wid = tid >> 5;                       // wave32 id, 0..7
  const long long row = blockIdx.x;               // (b*H + h)*N + r
  const int h = (int)((row >> 11) & 3);           // N = 2048, H = 4
  // K_LIST = {10, 20, 40, 500} == 10<<h for h<3; pure ALU, no constant pool
  const unsigned K = (h == 3) ? 500u : (10u << h);

  const float* __restrict__ rp = att + row * (long long)NCOL;
  float* __restrict__ op = out + row * (long long)NCOL;

  // ---- async stage: global -> LDS, 32 bytes per lane via two b128 copies ----
  {
    unsigned lds0 = (unsigned)(size_t)(&sdata[tid * EPT]);  // low 32 bits = LDS offset
    const float* g0 = rp + tid * EPT;
    asm volatile(
        "global_load_async_to_lds_b128 %0, %1, off\n\t"
        "global_load_async_to_lds_b128 %2, %3, off"
        :
        : "v"(lds0), "v"(g0), "v"(lds0 + 16u), "v"(g0 + 4)
        : "memory");
  }

  if (tid == 0) { s_maxkey = 0u; s_prefix = 0u; s_remk = K; }

  asm volatile("s_wait_asynccnt 0x0" ::: "memory");  // my wave's copies landed
  __syncthreads();                                    // whole tile visible

  // ---- pull my 8 elements into registers; keys + row max ----
  float fv[EPT];
  unsigned kv[EPT];
  {
    v4f a = *(const v4f*)&sdata[tid * EPT];
    v4f b = *(const v4f*)&sdata[tid * EPT + 4];
    fv[0]=a.x; fv[1]=a.y; fv[2]=a.z; fv[3]=a.w;
    fv[4]=b.x; fv[5]=b.y; fv[6]=b.z; fv[7]=b.w;
  }
  unsigned lmax = 0u;
  #pragma unroll
  for (int j = 0; j < EPT; ++j) {
    kv[j] = f2key(fv[j]);
    lmax = (kv[j] > lmax) ? kv[j] : lmax;
  }
  atomicMax(&s_maxkey, lmax);  // ds_max_u32

  unsigned* hflat = &shist[0][0];

  // ---- 4-pass MSB-first radix select: exact k-th largest key ----
  #pragma unroll
  for (int pass = 0; pass < 4; ++pass) {
    const int shift = 24 - pass * 8;
    #pragma unroll
    for (int j = 0; j < 8; ++j) hflat[tid + 256 * j] = 0u;   // clear 8 copies
    __syncthreads();

    const unsigned pref  = s_prefix;
    const unsigned pmask = (pass == 0) ? 0u : (0xFFFFFFFFu << (shift + 8));
    #pragma unroll
    for (int j = 0; j < EPT; ++j) {
      if ((kv[j] & pmask) == pref)
        atomicAdd(&shist[wid][(kv[j] >> shift) & 255u], 1u);
    }
    __syncthreads();

    // fold the 8 per-wave copies (conflict-free: lane-consecutive bins)
    unsigned c = 0;
    #pragma unroll
    for (int w = 0; w < 8; ++w) c += shist[w][tid];
    sred[tid] = c;
    __syncthreads();

    if (tid == 0) {
      const unsigned remk = s_remk;
      unsigned cum = 0, binc = 0;
      int d;
      for (d = 255; d >= 0; --d) {        // early exit: target bin is near the top
        binc = sred[d];
        cum += binc;
        if (cum >= remk) break;
      }
      if (d < 0) d = 0;                   // unreachable safety
      s_remk   = remk - (cum - binc);     // rank within the chosen bin
      s_prefix = s_prefix | ((unsigned)d << shift);
    }
    __syncthreads();
  }

  // ---- softmax over the selected set {key >= T} (exactly k elems, no ties) ----
  const unsigned T = s_prefix;
  const float maxf = key2f(s_maxkey);

  float local = 0.0f;
  #pragma unroll
  for (int j = 0; j < EPT; ++j)
    if (kv[j] >= T) local += __expf(fv[j] - maxf);

  sredf[tid] = local;
  __syncthreads();
  #pragma unroll
  for (int s = 128; s > 0; s >>= 1) {     // deterministic tree reduction
    if (tid < s) sredf[tid] += sredf[tid + s];
    __syncthreads();
  }
  const float inv = 1.0f / sredf[0];

  // ---- streamed NT output: exp(x-max)/sum at top-k positions, 0 elsewhere ----
  float o[EPT];
  #pragma unroll
  for (int j = 0; j < EPT; ++j)
    o[j] = (kv[j] >= T) ? __expf(fv[j] - maxf) * inv : 0.0f;

  v4f o0 = {o[0], o[1], o[2], o[3]};
  v4f o1 = {o[4], o[5], o[6], o[7]};
  __builtin_nontemporal_store(o0, (v4f*)(op + tid * EPT));
  __builtin_nontemporal_store(o1, (v4f*)(op + tid * EPT + 4));
}

extern "C" void kernel_launch(void* const* d_in, const int* in_sizes, int n_in,
                              void* d_out, int out_size, void* d_ws, size_t ws_size,
                              hipStream_t stream) {
  (void)n_in; (void)out_size; (void)d_ws; (void)ws_size;
  const float* att = (const float*)d_in[0];
  float* out = (float*)d_out;
  const int rows = in_sizes[0] / NCOL;  // 8*4*2048 = 65536
  hipLaunchKernelGGL(topk_softmax_kernel, dim3(rows), dim3(TPB), 0, stream,
                     att, out);
}